// SpatialRelationAttention_49924699848997
// MI455X (gfx1250) — compile-verified
//
#include <hip/hip_runtime.h>

#define B_DIM 8
#define C_DIM 256
#define HW 4096
#define LN_EPS 1e-5f

typedef __attribute__((ext_vector_type(2))) float v2f;
typedef __attribute__((ext_vector_type(8))) float v8f;

// ---------------------------------------------------------------------------
// Kernel 1: s[b,c] = sum_p k[b,c,p] * w_lin[p]   (streams k: 32 MB, coalesced)
// grid (C_DIM, B_DIM) x 256 threads
// ---------------------------------------------------------------------------
__global__ void sra_reduce_k(const float* __restrict__ k,
                             const float* __restrict__ wlin,
                             float* __restrict__ s_out) {
  const int c = blockIdx.x, b = blockIdx.y;
  const float4* kp = (const float4*)(k + ((size_t)(b * C_DIM + c)) * HW);
  const float4* wp = (const float4*)wlin;
  float acc = 0.f;
  for (int i = threadIdx.x; i < HW / 4; i += 256) {
    float4 kv = kp[i], wv = wp[i];
    acc += kv.x * wv.x + kv.y * wv.y + kv.z * wv.z + kv.w * wv.w;
  }
  __shared__ float red[256];
  red[threadIdx.x] = acc;
  __syncthreads();
  for (int off = 128; off > 0; off >>= 1) {
    if (threadIdx.x < off) red[threadIdx.x] += red[threadIdx.x + off];
    __syncthreads();
  }
  if (threadIdx.x == 0) s_out[b * C_DIM + c] = red[0];
}

// ---------------------------------------------------------------------------
// Kernel 2: the two surviving GEMMs via V_WMMA_F32_16X16X4_F32.
//   KV[m][n] = sum_k wk[m][k] * S[k][n] + bk[m]*Wsum     (M=256,N=16,K=256)
//   U [m][n] = sum_k wq[k][m] * KV[k][n]                 (M=256,N=16,K=256)
//   beta_b   = sum_o bq[o]*KV[o][b] + b_lin
// one workgroup, 256 threads = 8 wave32 waves, 2 M-tiles per wave
// ---------------------------------------------------------------------------
__global__ void sra_mix(const float* __restrict__ wq,
                        const float* __restrict__ bq,
                        const float* __restrict__ wk,
                        const float* __restrict__ bk,
                        const float* __restrict__ wlin,
                        const float* __restrict__ blin,
                        const float* __restrict__ s_in,
                        float* __restrict__ u_out,
                        float* __restrict__ beta_out) {
  __shared__ float sS[C_DIM * 16];   // S padded to N=16 (batches 8..15 = 0)
  __shared__ float sKV[C_DIM * 16];  // KV padded to N=16
  __shared__ float red[256];
  __shared__ float sWsum;

  const int tid = threadIdx.x;
  const int wave = tid >> 5;
  const int lane = tid & 31;
  const int lane_m = lane & 15;  // N index for B/C/D frags, M offset for A frag
  const int khalf = lane >> 4;   // K-pair select for A/B frags (ISA 16x4 layout)

  // Wsum = sum(w_lin)
  float acc0 = 0.f;
  for (int i = tid; i < HW; i += 256) acc0 += wlin[i];
  red[tid] = acc0;
  __syncthreads();
  for (int off = 128; off > 0; off >>= 1) {
    if (tid < off) red[tid] += red[tid + off];
    __syncthreads();
  }
  if (tid == 0) sWsum = red[0];

  // Stage S[k][n] into LDS, zero-padded to 16 columns
  for (int idx = tid; idx < C_DIM * 16; idx += 256) {
    int c = idx >> 4, n = idx & 15;
    sS[idx] = (n < B_DIM) ? s_in[n * C_DIM + c] : 0.f;
  }
  __syncthreads();

  // GEMM1: KV = wk @ S   (A = wk rows, row-major; B = S from LDS)
  for (int t = wave; t < 16; t += 8) {
    const int m0 = t * 16;
    const int m = m0 + lane_m;
    v8f acc = {};
#pragma unroll 4
    for (int kk = 0; kk < C_DIM; kk += 4) {
      const int kx = kk + 2 * khalf;
      v2f a, bmat;
      a.x = wk[m * C_DIM + kx];          // A 16x4: lanes<16 K=0,1 ; lanes>=16 K=2,3
      a.y = wk[m * C_DIM + kx + 1];
      bmat.x = sS[kx * 16 + lane_m];     // B 4x16: same K split, N = lane&15
      bmat.y = sS[(kx + 1) * 16 + lane_m];
      acc = __builtin_amdgcn_wmma_f32_16x16x4_f32(false, a, false, bmat,
                                                  (short)0, acc, false, false);
    }
    const float wsum = sWsum;
#pragma unroll
    for (int v = 0; v < 8; v++) {        // D: VGPR v -> M = v (lanes<16) / v+8
      const int row = m0 + v + 8 * khalf;
      sKV[row * 16 + lane_m] = acc[v] + bk[row] * wsum;
    }
  }
  __syncthreads();

  // beta_b = bq . KV[:,b] + b_lin
  if (tid < B_DIM) {
    float accb = 0.f;
    for (int o = 0; o < C_DIM; o++) accb += bq[o] * sKV[o * 16 + tid];
    beta_out[tid] = accb + blin[0];
  }

  // GEMM2: U = wq^T @ KV   (A fetch wq[k][m] is coalesced across lanes)
  for (int t = wave; t < 16; t += 8) {
    const int m0 = t * 16;
    const int m = m0 + lane_m;
    v8f acc = {};
#pragma unroll 4
    for (int kk = 0; kk < C_DIM; kk += 4) {
      const int kx = kk + 2 * khalf;
      v2f a, bmat;
      a.x = wq[kx * C_DIM + m];
      a.y = wq[(kx + 1) * C_DIM + m];
      bmat.x = sKV[kx * 16 + lane_m];
      bmat.y = sKV[(kx + 1) * 16 + lane_m];
      acc = __builtin_amdgcn_wmma_f32_16x16x4_f32(false, a, false, bmat,
                                                  (short)0, acc, false, false);
    }
    if (lane_m < B_DIM) {                // store only the 8 real batch columns
#pragma unroll
      for (int v = 0; v < 8; v++) {
        const int row = m0 + v + 8 * khalf;
        u_out[lane_m * C_DIM + row] = acc[v];
      }
    }
  }
}

// ---------------------------------------------------------------------------
// Kernel 3: val[b,p] = sum_c q[b,c,p] * u[b,c] + beta_b  (streams q: 32 MB)
// grid (16, B_DIM) x 256 threads; each c-iteration loads 1KB contiguous/block
// ---------------------------------------------------------------------------
__global__ void sra_val(const float* __restrict__ q,
                        const float* __restrict__ u,
                        const float* __restrict__ beta_b,
                        float* __restrict__ val) {
  const int b = blockIdx.y;
  const int p = blockIdx.x * 256 + threadIdx.x;
  __shared__ float su[C_DIM];
  su[threadIdx.x] = u[b * C_DIM + threadIdx.x];
  __syncthreads();
  const float* qp = q + (size_t)b * C_DIM * HW + p;
  float acc = beta_b[b];
#pragma unroll 8
  for (int c = 0; c < C_DIM; c++) acc += qp[(size_t)c * HW] * su[c];
  val[b * HW + p] = acc;
}

// ---------------------------------------------------------------------------
// Kernel 4: LayerNorm over HW per batch (biased var, matches jnp.var)
// grid (B_DIM) x 256 threads
// ---------------------------------------------------------------------------
__global__ void sra_ln(const float* __restrict__ val,
                       const float* __restrict__ gamma,
                       const float* __restrict__ beta_ln,
                       float* __restrict__ out) {
  const int b = blockIdx.x;
  const float* v = val + b * HW;
  float s = 0.f, ss = 0.f;
  for (int i = threadIdx.x; i < HW; i += 256) {
    float x = v[i];
    s += x;
    ss += x * x;
  }
  __shared__ float rs[256], rss[256];
  rs[threadIdx.x] = s;
  rss[threadIdx.x] = ss;
  __syncthreads();
  for (int off = 128; off > 0; off >>= 1) {
    if (threadIdx.x < off) {
      rs[threadIdx.x] += rs[threadIdx.x + off];
      rss[threadIdx.x] += rss[threadIdx.x + off];
    }
    __syncthreads();
  }
  __shared__ float smean, srstd;
  if (threadIdx.x == 0) {
    float mean = rs[0] * (1.f / HW);
    float var = rss[0] * (1.f / HW) - mean * mean;
    smean = mean;
    srstd = rsqrtf(var + LN_EPS);
  }
  __syncthreads();
  const float mean = smean, rstd = srstd;
  for (int i = threadIdx.x; i < HW; i += 256) {
    float x = v[i];
    out[b * HW + i] = (x - mean) * rstd * gamma[i] + beta_ln[i];
  }
}

// ---------------------------------------------------------------------------
extern "C" void kernel_launch(void* const* d_in, const int* in_sizes, int n_in,
                              void* d_out, int out_size, void* d_ws, size_t ws_size,
                              hipStream_t stream) {
  const float* q     = (const float*)d_in[0];
  const float* k     = (const float*)d_in[1];
  const float* wq    = (const float*)d_in[2];
  const float* bq    = (const float*)d_in[3];
  const float* wk    = (const float*)d_in[4];
  const float* bk    = (const float*)d_in[5];
  const float* wlin  = (const float*)d_in[6];
  const float* blin  = (const float*)d_in[7];
  const float* gamma = (const float*)d_in[8];
  const float* betaL = (const float*)d_in[9];
  float* out = (float*)d_out;

  float* ws       = (float*)d_ws;
  float* s_buf    = ws;          // [B, C]  = 2048 floats
  float* u_buf    = ws + 2048;   // [B, C]  = 2048 floats
  float* beta_buf = ws + 4096;   // [B]     (padded)
  float* val_buf  = ws + 4160;   // [B, HW] = 32768 floats

  sra_reduce_k<<<dim3(C_DIM, B_DIM), 256, 0, stream>>>(k, wlin, s_buf);
  sra_mix<<<1, 256, 0, stream>>>(wq, bq, wk, bk, wlin, blin, s_buf, u_buf, beta_buf);
  sra_val<<<dim3(HW / 256, B_DIM), 256, 0, stream>>>(q, u_buf, beta_buf, val_buf);
  sra_ln<<<B_DIM, 256, 0, stream>>>(val_buf, gamma, betaL, out);
}